// MemoryWriter_61194694033988
// MI455X (gfx1250) — compile-verified
//
#include <hip/hip_runtime.h>

#define N_SLOTS 65536
#define DIM 128
#define BATCH 4096
#define TOPK 8
#define UPDATE_RATE 0.1f
#define MOMENTUM 0.9f
#define GATE_THRESH 0.01f

#define ACC_ELEMS ((size_t)N_SLOTS * DIM)   // 8,388,608 floats = 32 MB

typedef float f32x4 __attribute__((ext_vector_type(4)));

// ---------------------------------------------------------------------------
// Kernel 1: zero the workspace (key_acc | val_acc | counts), 128-bit wide.
// Regular-temporal stores: we WANT these lines resident in the 192 MB L2 so
// the scatter atomics and finalize reads hit on-chip.
// ---------------------------------------------------------------------------
__global__ void __launch_bounds__(256) zero_ws_kernel(f32x4* __restrict__ p,
                                                      unsigned n4) {
  unsigned i = blockIdx.x * 256u + threadIdx.x;
  if (i < n4) p[i] = (f32x4){0.f, 0.f, 0.f, 0.f};
}

// ---------------------------------------------------------------------------
// Kernel 2: gated scatter-accumulate. One wave32 per write row: each lane
// owns a float4 slice of the 128-wide row; 4 f32 atomics per matrix per slot.
// The 64 MB accumulator region stays L2-resident, so atomics resolve on-chip.
// row is wave-uniform -> readfirstlane so gate/idx loads become scalar SMEM.
// Query/value rows are read exactly once -> non-temporal loads.
// ---------------------------------------------------------------------------
__global__ void __launch_bounds__(256) scatter_kernel(
    const float* __restrict__ q, const float* __restrict__ wv,
    const float* __restrict__ gate, const int* __restrict__ idx,
    float* __restrict__ kacc, float* __restrict__ vacc,
    float* __restrict__ cnt) {
  unsigned gtid = blockIdx.x * 256u + threadIdx.x;
  unsigned lane = gtid & 31u;
  unsigned row  = __builtin_amdgcn_readfirstlane(gtid >> 5);  // wave-uniform
  if (row >= BATCH) return;

  float g = gate[row];
  float w = (g > GATE_THRESH) ? g * UPDATE_RATE : 0.f;
  if (w == 0.f) return;           // uniform across the wave

  f32x4 qv = __builtin_nontemporal_load(
      (const f32x4*)(q + (size_t)row * DIM) + lane);
  f32x4 vv = __builtin_nontemporal_load(
      (const f32x4*)(wv + (size_t)row * DIM) + lane);
  qv *= w;
  vv *= w;

#pragma unroll
  for (int k = 0; k < TOPK; ++k) {
    int s = idx[row * TOPK + k];                 // scalar load (uniform addr)
    float* kd = kacc + (size_t)s * DIM + lane * 4u;
    float* vd = vacc + (size_t)s * DIM + lane * 4u;
    atomicAdd(kd + 0, qv.x); atomicAdd(kd + 1, qv.y);
    atomicAdd(kd + 2, qv.z); atomicAdd(kd + 3, qv.w);
    atomicAdd(vd + 0, vv.x); atomicAdd(vd + 1, vv.y);
    atomicAdd(vd + 2, vv.z); atomicAdd(vd + 3, vv.w);
    if (lane == 0) atomicAdd(cnt + s, w);
  }
}

// ---------------------------------------------------------------------------
// Kernel 3: finalize. out = mem + 0.9*mom + (0.1/denom)*acc, 128-bit streams.
//  - memory/momentum loads and output stores: non-temporal (one-touch streams,
//    must not evict the L2-resident accumulators)
//  - accumulator loads: regular temporal (expected L2 hits)
//  - per-block counts tile (8 slots): staged into LDS with the gfx1250 async
//    global->LDS path + s_wait_asynccnt (split CDNA5 wait counters).
// LDS byte offset = low 32 bits of the generic shared pointer.
// ---------------------------------------------------------------------------
__device__ __forceinline__ f32x4 mix4(f32x4 m, f32x4 mo, f32x4 a, float s) {
  f32x4 o;
  o.x = fmaf(s, a.x, fmaf(MOMENTUM, mo.x, m.x));
  o.y = fmaf(s, a.y, fmaf(MOMENTUM, mo.y, m.y));
  o.z = fmaf(s, a.z, fmaf(MOMENTUM, mo.z, m.z));
  o.w = fmaf(s, a.w, fmaf(MOMENTUM, mo.w, m.w));
  return o;
}

__global__ void __launch_bounds__(256) finalize_kernel(
    const f32x4* __restrict__ mk, const f32x4* __restrict__ mv,
    const f32x4* __restrict__ km, const f32x4* __restrict__ vm,
    const f32x4* __restrict__ kacc, const f32x4* __restrict__ vacc,
    const float* __restrict__ cnt,
    f32x4* __restrict__ outk, f32x4* __restrict__ outv) {
  __shared__ float scnt[8];

  unsigned i     = blockIdx.x * 256u + threadIdx.x;  // float4 index
  unsigned slot0 = blockIdx.x * 8u;                  // 256 thr * 4 f32 = 8 slots

  if (threadIdx.x < 8u) {
    unsigned lds_off = (unsigned)(unsigned long long)&scnt[threadIdx.x];
    const float* gp  = cnt + slot0 + threadIdx.x;
    asm volatile("global_load_async_to_lds_b32 %0, %1, off"
                 :: "v"(lds_off), "v"(gp) : "memory");
    asm volatile("s_wait_asynccnt 0" ::: "memory");
  }
  __syncthreads();

  float c     = scnt[threadIdx.x >> 5];
  float denom = (c > 0.f) ? c : 1.f;
  float s     = (1.0f - MOMENTUM) / denom;

  f32x4 mk4 = __builtin_nontemporal_load(mk + i);
  f32x4 km4 = __builtin_nontemporal_load(km + i);
  f32x4 ka4 = kacc[i];                               // L2-resident, RT
  __builtin_nontemporal_store(mix4(mk4, km4, ka4, s), outk + i);

  f32x4 mv4 = __builtin_nontemporal_load(mv + i);
  f32x4 vm4 = __builtin_nontemporal_load(vm + i);
  f32x4 va4 = vacc[i];                               // L2-resident, RT
  __builtin_nontemporal_store(mix4(mv4, vm4, va4, s), outv + i);
}

// ---------------------------------------------------------------------------
// Host launch: ws layout = [key_acc 32MB | val_acc 32MB | counts 256KB].
// ---------------------------------------------------------------------------
extern "C" void kernel_launch(void* const* d_in, const int* in_sizes, int n_in,
                              void* d_out, int out_size, void* d_ws, size_t ws_size,
                              hipStream_t stream) {
  (void)in_sizes; (void)n_in; (void)out_size; (void)ws_size;

  const float* mk  = (const float*)d_in[0];
  const float* mv  = (const float*)d_in[1];
  const float* q   = (const float*)d_in[2];
  const float* wv  = (const float*)d_in[3];
  const float* g   = (const float*)d_in[4];
  const int*   idx = (const int*)  d_in[5];
  const float* km  = (const float*)d_in[6];
  const float* vm  = (const float*)d_in[7];

  float* out  = (float*)d_out;
  float* kacc = (float*)d_ws;
  float* vacc = kacc + ACC_ELEMS;
  float* cnt  = vacc + ACC_ELEMS;

  // 1) zero accumulators + counts (16,842,752 floats, divisible by 4)
  unsigned n4 = (unsigned)((2 * ACC_ELEMS + N_SLOTS) / 4);
  zero_ws_kernel<<<(n4 + 255u) / 256u, 256, 0, stream>>>((f32x4*)d_ws, n4);

  // 2) scatter-accumulate: one wave32 per write row
  scatter_kernel<<<(BATCH * 32) / 256, 256, 0, stream>>>(q, wv, g, idx,
                                                         kacc, vacc, cnt);

  // 3) finalize: 2M float4 threads, 8 slots per 256-thread block
  unsigned nf4 = (unsigned)(ACC_ELEMS / 4);
  finalize_kernel<<<nf4 / 256u, 256, 0, stream>>>(
      (const f32x4*)mk, (const f32x4*)mv,
      (const f32x4*)km, (const f32x4*)vm,
      (const f32x4*)kacc, (const f32x4*)vacc, cnt,
      (f32x4*)out, (f32x4*)(out + ACC_ELEMS));
}